// LapResNet2_3496103379445
// MI455X (gfx1250) — compile-verified
//
#include <hip/hip_runtime.h>
#include <math.h>

// ---------------------------------------------------------------------------
// LapResNet2 forward for MI455X (gfx1250), fp32 end-to-end using
// V_WMMA_F32_16X16X4_F32 for all GEMMs (dtype-exact vs the fp32 reference).
//
// gfx1250-specific paths:
//   - v_wmma_f32_16x16x4_f32 for all matrix math
//   - global_load_async_to_lds_b128 (ASYNCcnt) double-buffered A-tile stream
//   - s_wait_asynccnt fencing
// ---------------------------------------------------------------------------

typedef __attribute__((ext_vector_type(2))) float v2f;
typedef __attribute__((ext_vector_type(4))) float f4;
typedef __attribute__((ext_vector_type(8))) float v8f;

#define NROWS   8192            // BATCH*NODE
#define DIMF    128
#define TWODIM  256
#define EPSBN   1e-5f

#define BM 128
#define BN 128
#define BK 32
#define ASTRIDE 36              // A tile LDS stride (mult of 4 for b128, bank-spread)
#define BPT     34              // transposed-B tile LDS stride (even for b64 frags)

__device__ __forceinline__ float elu_f(float x) {
    return x > 0.0f ? x : expm1f(x);
}

// async DMA: 16 bytes global -> LDS, tracked by ASYNCcnt (no VGPR staging)
__device__ __forceinline__ void async_b128(unsigned lds_byte_addr, unsigned voff_bytes,
                                           unsigned long long sbase) {
    asm volatile("global_load_async_to_lds_b128 %0, %1, %2"
                 :: "v"(lds_byte_addr), "v"(voff_bytes), "s"(sbase)
                 : "memory");
}
__device__ __forceinline__ void wait_async0() {
    asm volatile("s_wait_asynccnt 0x0" ::: "memory");
}

// -------------------------------- ELU ------------------------------------
__global__ __launch_bounds__(256) void elu_kernel(const float* __restrict__ in,
                                                  float* __restrict__ out, int n) {
    int i = blockIdx.x * blockDim.x + threadIdx.x;
    if (i < n) out[i] = elu_f(in[i]);
}

// ------------------------------ zero stats --------------------------------
__global__ __launch_bounds__(256) void zero_kernel(float* __restrict__ p, int n) {
    int i = blockIdx.x * blockDim.x + threadIdx.x;
    if (i < n) p[i] = 0.0f;
}

// ------------------- big GEMM: Y(8192x128) = L @ X ------------------------
// grid.x = 64 (M tiles of 128), block = 256 threads = 8 waves.
// A (L tile, 128x32) streams via async-to-LDS, double buffered.
// B (X tile, 32x128) is register-staged and stored TRANSPOSED (Bst[n][k]) so a
// WMMA B fragment is one contiguous ds_load_b64 per lane (no v_mov regrouping).
__global__ __launch_bounds__(256) void gemm_L_kernel(const float* __restrict__ L,
                                                     const float* __restrict__ X,
                                                     float* __restrict__ Y) {
    __shared__ __align__(16) float As[2][BM * ASTRIDE];
    __shared__ __align__(16) float Bst[DIMF * BPT];

    const int tid   = threadIdx.x;
    const int wave  = tid >> 5;
    const int lane  = tid & 31;
    const int rowM  = blockIdx.x * BM;
    const int mBand = wave * 16;
    const int lrow  = lane & 15;
    const int khalf = (lane < 16) ? 0 : 2;   // ISA 16x4 f32 lane->K mapping

    v8f acc[8] = {};

    // A: 128x32 tile, one b128 per thread x4 ; B: 32x128 tile, one f4 x4
    const int a_c4 = (tid & 7) * 4;          // 0..28
    const int a_r  = tid >> 3;               // 0..31
    const int b_c4 = (tid & 31) * 4;         // 0..124
    const int b_r  = tid >> 5;               // 0..7

    // loop-invariant per-lane byte offsets for the async A stream (GVS mode:
    // sgpr base advances by BK floats each K-step, voffset stays fixed)
    unsigned aVoff[4], aLds[2][4];
#pragma unroll
    for (int i = 0; i < 4; ++i) {
        aVoff[i]   = 4u * ((unsigned)(rowM + a_r + 32 * i) * (unsigned)NROWS
                           + (unsigned)a_c4);
        aLds[0][i] = (unsigned)(size_t)&As[0][(a_r + 32 * i) * ASTRIDE + a_c4];
        aLds[1][i] = (unsigned)(size_t)&As[1][(a_r + 32 * i) * ASTRIDE + a_c4];
    }

    f4 b_reg[4];

    // prologue: async A tile 0 -> As[0]; B tile 0 -> regs
#pragma unroll
    for (int i = 0; i < 4; ++i)
        async_b128(aLds[0][i], aVoff[i], (unsigned long long)L);
#pragma unroll
    for (int i = 0; i < 4; ++i)
        b_reg[i] = *(const f4*)(X + (size_t)(b_r + 8 * i) * DIMF + b_c4);

    for (int kb = 0; kb < NROWS; kb += BK) {
        const int cur = (kb >> 5) & 1;

        __syncthreads();               // all waves done reading Bst (prev tile)
#pragma unroll
        for (int i = 0; i < 4; ++i)
#pragma unroll
            for (int j = 0; j < 4; ++j)
                Bst[(b_c4 + j) * BPT + (b_r + 8 * i)] = b_reg[i][j];

        wait_async0();                 // A tile `cur` landed in LDS
        __syncthreads();               // A + B tiles visible to all waves

        if (kb + BK < NROWS) {         // overlap next tile with compute
#pragma unroll
            for (int i = 0; i < 4; ++i)
                async_b128(aLds[cur ^ 1][i], aVoff[i],
                           (unsigned long long)(L + kb + BK));
#pragma unroll
            for (int i = 0; i < 4; ++i)
                b_reg[i] = *(const f4*)(X + (size_t)(kb + BK + b_r + 8 * i) * DIMF + b_c4);
        }

#pragma unroll
        for (int ks = 0; ks < BK; ks += 4) {
            v2f a = *(const v2f*)&As[cur][(mBand + lrow) * ASTRIDE + ks + khalf];
#pragma unroll
            for (int nt = 0; nt < 8; ++nt) {
                v2f b = *(const v2f*)&Bst[(nt * 16 + lrow) * BPT + ks + khalf];
                acc[nt] = __builtin_amdgcn_wmma_f32_16x16x4_f32(
                    false, a, false, b, (short)0, acc[nt], false, false);
            }
        }
    }

    // C/D layout: VGPR i -> M = i (lanes 0-15) / 8+i (lanes 16-31), N = lane%16
    const int rOut = rowM + mBand + ((lane < 16) ? 0 : 8);
#pragma unroll
    for (int nt = 0; nt < 8; ++nt)
#pragma unroll
        for (int i = 0; i < 8; ++i)
            Y[(size_t)(rOut + i) * DIMF + nt * 16 + lrow] = acc[nt][i];
}

// -------------------- per-column sum / sumsq of [x | Lx] ------------------
__global__ __launch_bounds__(256) void colstats_kernel(const float* __restrict__ X,
                                                       const float* __restrict__ Lx,
                                                       float* __restrict__ colsum,
                                                       float* __restrict__ colsumsq) {
    const int tid  = threadIdx.x;
    const float* src = (tid < DIMF) ? X : Lx;
    const int col  = tid & (DIMF - 1);
    const int row0 = blockIdx.x * 128;
    float s = 0.0f, ss = 0.0f;
    for (int r = 0; r < 128; ++r) {
        float v = src[(size_t)(row0 + r) * DIMF + col];
        s += v;
        ss += v * v;
    }
    atomicAdd(&colsum[tid], s);
    atomicAdd(&colsumsq[tid], ss);
}

// ---- fold BN into FC:  BT[k][n] = w[n][k]*s[k],  cvec[n] = b[n]+sum t[k]w[n][k]
__global__ __launch_bounds__(256) void prep_kernel(const float* __restrict__ colsum,
                                                   const float* __restrict__ colsumsq,
                                                   const float* __restrict__ gamma,
                                                   const float* __restrict__ beta,
                                                   const float* __restrict__ w,   // (128 x 256)
                                                   const float* __restrict__ bias,
                                                   float* __restrict__ BT,        // (256 x 128)
                                                   float* __restrict__ cvec) {    // (128)
    __shared__ float s_s[TWODIM];
    __shared__ float s_t[TWODIM];
    const int tid = threadIdx.x;
    const float inv_n = 1.0f / (float)NROWS;
    {
        float mu  = colsum[tid] * inv_n;
        float var = colsumsq[tid] * inv_n - mu * mu;
        float sc  = gamma[tid] * rsqrtf(var + EPSBN);
        s_s[tid] = sc;
        s_t[tid] = beta[tid] - mu * sc;
    }
    __syncthreads();
    for (int idx = tid; idx < TWODIM * DIMF; idx += 256) {
        int k = idx >> 7;
        int n = idx & 127;
        BT[idx] = w[n * TWODIM + k] * s_s[k];
    }
    if (tid < DIMF) {
        float c = bias[tid];
        for (int k = 0; k < TWODIM; ++k) c += s_t[k] * w[tid * TWODIM + k];
        cvec[tid] = c;
    }
}

// ---- FC GEMM: out(8192x128) = [X|Lx](8192x256) @ BT(256x128) + cvec, epilogue
__global__ __launch_bounds__(256) void gemm_fc_kernel(const float* __restrict__ X,
                                                      const float* __restrict__ Lx,
                                                      const float* __restrict__ BT,
                                                      const float* __restrict__ cvec,
                                                      const float* __restrict__ residual,
                                                      float* __restrict__ out,
                                                      int do_elu) {
    __shared__ __align__(16) float As[BM * ASTRIDE];
    __shared__ __align__(16) float Bst[DIMF * BPT];

    const int tid   = threadIdx.x;
    const int wave  = tid >> 5;
    const int lane  = tid & 31;
    const int rowM  = blockIdx.x * BM;
    const int mBand = wave * 16;
    const int lrow  = lane & 15;
    const int khalf = (lane < 16) ? 0 : 2;

    v8f acc[8] = {};

    const int a_c4 = (tid & 7) * 4;
    const int a_r  = tid >> 3;
    const int b_c4 = (tid & 31) * 4;
    const int b_r  = tid >> 5;

    for (int kb = 0; kb < TWODIM; kb += BK) {
        const float* srcA = (kb < DIMF) ? X : Lx;   // concat [x | Lx] split on BK boundary
        const int colb = kb & (DIMF - 1);
        f4 a_reg[4], b_reg[4];
#pragma unroll
        for (int i = 0; i < 4; ++i) {
            int r = a_r + 32 * i;
            a_reg[i] = *(const f4*)(srcA + (size_t)(rowM + r) * DIMF + colb + a_c4);
        }
#pragma unroll
        for (int i = 0; i < 4; ++i) {
            int r = b_r + 8 * i;
            b_reg[i] = *(const f4*)(BT + (size_t)(kb + r) * DIMF + b_c4);
        }
        __syncthreads();
#pragma unroll
        for (int i = 0; i < 4; ++i)
            *(f4*)&As[(a_r + 32 * i) * ASTRIDE + a_c4] = a_reg[i];
#pragma unroll
        for (int i = 0; i < 4; ++i)
#pragma unroll
            for (int j = 0; j < 4; ++j)
                Bst[(b_c4 + j) * BPT + (b_r + 8 * i)] = b_reg[i][j];
        __syncthreads();

#pragma unroll
        for (int ks = 0; ks < BK; ks += 4) {
            v2f a = *(const v2f*)&As[(mBand + lrow) * ASTRIDE + ks + khalf];
#pragma unroll
            for (int nt = 0; nt < 8; ++nt) {
                v2f b = *(const v2f*)&Bst[(nt * 16 + lrow) * BPT + ks + khalf];
                acc[nt] = __builtin_amdgcn_wmma_f32_16x16x4_f32(
                    false, a, false, b, (short)0, acc[nt], false, false);
            }
        }
    }

    const int rOut = rowM + mBand + ((lane < 16) ? 0 : 8);
#pragma unroll
    for (int nt = 0; nt < 8; ++nt) {
        const int col = nt * 16 + lrow;
        const float cadd = cvec[col];
#pragma unroll
        for (int i = 0; i < 8; ++i) {
            float v = acc[nt][i] + cadd;
            if (do_elu) v = elu_f(v);
            if (residual) v += residual[(size_t)(rOut + i) * DIMF + col];
            out[(size_t)(rOut + i) * DIMF + col] = v;
        }
    }
}

// ---------------------------------------------------------------------------
extern "C" void kernel_launch(void* const* d_in, const int* in_sizes, int n_in,
                              void* d_out, int out_size, void* d_ws, size_t ws_size,
                              hipStream_t stream) {
    const float* L         = (const float*)d_in[0];
    /* d_in[1] = mask (unused) */
    const float* inputs    = (const float*)d_in[2];
    const float* bn0_gamma = (const float*)d_in[3];
    const float* bn0_beta  = (const float*)d_in[4];
    const float* fc0_w     = (const float*)d_in[5];
    const float* fc0_b     = (const float*)d_in[6];
    const float* bn1_gamma = (const float*)d_in[7];
    const float* bn1_beta  = (const float*)d_in[8];
    const float* fc1_w     = (const float*)d_in[9];
    const float* fc1_b     = (const float*)d_in[10];
    float* out = (float*)d_out;

    // workspace layout (floats)
    float* ws = (float*)d_ws;
    const size_t NF = (size_t)NROWS * DIMF;      // 1,048,576
    float* x0      = ws;                         // elu(inputs)
    float* Lx      = x0 + NF;
    float* x1      = Lx + NF;
    float* colsum  = x1 + NF;                    // 256
    float* colsq   = colsum + TWODIM;            // 256
    float* BT      = colsq + TWODIM;             // 256*128
    float* cvec    = BT + (size_t)TWODIM * DIMF; // 128

    const int nElems = NROWS * DIMF;
    const dim3 blk(256);

    // ---- block 0 ----
    elu_kernel<<<(nElems + 255) / 256, blk, 0, stream>>>(inputs, x0, nElems);
    zero_kernel<<<2, blk, 0, stream>>>(colsum, 2 * TWODIM);
    gemm_L_kernel<<<NROWS / BM, blk, 0, stream>>>(L, x0, Lx);
    colstats_kernel<<<NROWS / 128, blk, 0, stream>>>(x0, Lx, colsum, colsq);
    prep_kernel<<<1, blk, 0, stream>>>(colsum, colsq, bn0_gamma, bn0_beta,
                                       fc0_w, fc0_b, BT, cvec);
    gemm_fc_kernel<<<NROWS / BM, blk, 0, stream>>>(x0, Lx, BT, cvec,
                                                   nullptr, x1, /*do_elu=*/1);

    // ---- block 1 ----
    zero_kernel<<<2, blk, 0, stream>>>(colsum, 2 * TWODIM);
    gemm_L_kernel<<<NROWS / BM, blk, 0, stream>>>(L, x1, Lx);
    colstats_kernel<<<NROWS / 128, blk, 0, stream>>>(x1, Lx, colsum, colsq);
    prep_kernel<<<1, blk, 0, stream>>>(colsum, colsq, bn1_gamma, bn1_beta,
                                       fc1_w, fc1_b, BT, cvec);
    gemm_fc_kernel<<<NROWS / BM, blk, 0, stream>>>(x1, Lx, BT, cvec,
                                                   inputs, out, /*do_elu=*/0);
}